// AddAttention_49924699849072
// MI455X (gfx1250) — compile-verified
//
#include <hip/hip_runtime.h>
#include <hip/hip_bf16.h>
#include <math.h>

// Problem: B=16, L=2048, D=256
// out  = broadcast of ctx[b,:] over i   (16*2048*256 floats)
// attn = broadcast of p[b,:]   over i   (16*2048*2048 floats)

#define Bsz 16
#define Lsz 2048
#define Dsz 256

typedef __attribute__((ext_vector_type(2))) float v2f;
typedef __attribute__((ext_vector_type(4))) float v4f;
typedef __attribute__((ext_vector_type(8))) float v8f;

// ---------------------------------------------------------------------------
// Kernel 1: sk[b,l] = inp[b,l,:] . wk  via V_WMMA_F32_16X16X4_F32.
// One wave -> 16 rows of sk. A = 16x4 tile of inp rows, B = 4x16 with only
// column 0 holding wk[d0..d0+3]. 64 WMMA steps cover K=256.
// B is masked arithmetically (w * msk) so the wk load stays unconditional:
// no exec-mask branches, EXEC is all-1s throughout (WMMA requirement).
// Grid: 256 blocks x 256 threads (8 waves/block, 16 rows/wave -> 32768 rows).
// ---------------------------------------------------------------------------
__global__ void sk_wmma_kernel(const float* __restrict__ inp,
                               const float* __restrict__ wk,
                               float* __restrict__ sk) {
  const int tid  = threadIdx.x;
  const int wave = tid >> 5;
  const int lane = tid & 31;
  const int m    = lane & 15;   // A-matrix row owned by this lane
  const int half = lane >> 4;   // 0: K pair {0,1}, 1: K pair {2,3}

  const long rowTile = (long)blockIdx.x * 8 + wave; // 16-row tile id
  const long l0      = rowTile * 16;                // flat row in [0, B*L)
  const float* rowBase = inp + (l0 + (long)m) * Dsz + 2 * half;
  const float* wkBase  = wk + 2 * half;

  // Lane mask for the B operand: only column N==0 (lanes 0 and 16) is live.
  const float msk = (m == 0) ? 1.0f : 0.0f;

  v8f c = {};
  #pragma unroll 8
  for (int d0 = 0; d0 < Dsz; d0 += 4) {
    // A layout (16x4 f32): lane m, VGPR0 = A[m, 2*half], VGPR1 = A[m, 2*half+1]
    const v2f a = *(const v2f*)(rowBase + d0);
    // B layout (4x16 f32): lanes 0-15 hold rows K0/K1 (N = lane),
    // lanes 16-31 hold rows K2/K3 (N = lane-16). Only N==0 is nonzero.
    const v2f w = *(const v2f*)(wkBase + d0);   // unconditional broadcast load
    const v2f b = w * msk;                       // arithmetic mask, no branch
    c = __builtin_amdgcn_wmma_f32_16x16x4_f32(
        /*neg_a=*/false, a, /*neg_b=*/false, b,
        /*c_mod=*/(short)0, c, /*reuse_a=*/false, /*reuse_b=*/false);
  }
  // D column 0: lane 0 VGPR r = D[M=r,0], lane 16 VGPR r = D[M=8+r,0]
  if (m == 0) {
    float* o = sk + l0 + half * 8;
    #pragma unroll
    for (int r = 0; r < 8; ++r) o[r] = c[r];
  }
}

// ---------------------------------------------------------------------------
// Kernel 2: per-batch softmax over 2048 sk values -> p[b,:]
// Grid: 16 blocks x 256 threads.
// ---------------------------------------------------------------------------
__global__ void softmax_kernel(const float* __restrict__ sk,
                               float* __restrict__ p) {
  __shared__ float red[256];
  const int b = blockIdx.x, tid = threadIdx.x;
  const float* s = sk + b * Lsz;

  float local[8];
  float mx = -INFINITY;
  #pragma unroll
  for (int k = 0; k < 8; ++k) {
    local[k] = s[tid + k * 256];
    mx = fmaxf(mx, local[k]);
  }
  red[tid] = mx; __syncthreads();
  for (int off = 128; off > 0; off >>= 1) {
    if (tid < off) red[tid] = fmaxf(red[tid], red[tid + off]);
    __syncthreads();
  }
  mx = red[0]; __syncthreads();

  float e[8], sum = 0.0f;
  #pragma unroll
  for (int k = 0; k < 8; ++k) { e[k] = __expf(local[k] - mx); sum += e[k]; }
  red[tid] = sum; __syncthreads();
  for (int off = 128; off > 0; off >>= 1) {
    if (tid < off) red[tid] += red[tid + off];
    __syncthreads();
  }
  const float inv = 1.0f / red[0];

  float* pp = p + b * Lsz;
  #pragma unroll
  for (int k = 0; k < 8; ++k) pp[tid + k * 256] = e[k] * inv;
}

// ---------------------------------------------------------------------------
// Kernel 3: split-K partial context. Block (b, chunk) accumulates
// partial[b,chunk,d] = sum_{j in chunk of 256} p[b,j]*inp[b,j,d].
// Deterministic (fixed loop order), no atomics. Grid: (16,8) x 256 threads.
// ---------------------------------------------------------------------------
__global__ void ctx_partial_kernel(const float* __restrict__ inp,
                                   const float* __restrict__ p,
                                   float* __restrict__ partial) {
  const int b = blockIdx.x, chunk = blockIdx.y, d = threadIdx.x;
  const float* pb   = p + b * Lsz + chunk * 256;
  const float* base = inp + ((size_t)b * Lsz + (size_t)chunk * 256) * Dsz + d;
  float acc = 0.0f;
  #pragma unroll 4
  for (int j = 0; j < 256; ++j)
    acc = fmaf(pb[j], base[(size_t)j * Dsz], acc);
  partial[(b * 8 + chunk) * Dsz + d] = acc;
}

// ---------------------------------------------------------------------------
// Kernel 4: broadcast writer (the real cost: ~300 MB of streaming stores).
// Block = (batch b, 16-row i-chunk). Stages p row + reduced ctx row in LDS,
// then blasts out/attn rows with non-temporal B128 stores (attn > L2 size).
// Grid: 2048 blocks x 256 threads.
// ---------------------------------------------------------------------------
__global__ void write_out_kernel(const float* __restrict__ p,
                                 const float* __restrict__ partial,
                                 float* __restrict__ out,
                                 float* __restrict__ attn) {
  const int b   = blockIdx.x >> 7;   // 0..15
  const int ic  = blockIdx.x & 127;  // 16-row chunk within batch
  const int tid = threadIdx.x;

  __shared__ __align__(16) float ctx[Dsz];
  __shared__ __align__(16) float prow[Lsz];

  // Final (deterministic) reduction of the 8 split-K partials.
  float acc = 0.0f;
  #pragma unroll
  for (int c = 0; c < 8; ++c) acc += partial[(b * 8 + c) * Dsz + tid];
  ctx[tid] = acc;

  // Stage p[b,:] (8 KB) in LDS.
  const v4f* p4  = (const v4f*)(p + b * Lsz);
  v4f*       pr4 = (v4f*)prow;
  #pragma unroll
  for (int k = 0; k < 2; ++k) pr4[tid + k * 256] = p4[tid + k * 256];
  __syncthreads();

  // out rows: 16 x 256 floats = 1024 v4f
  v4f* out4 = (v4f*)(out + ((size_t)b * Lsz + (size_t)ic * 16) * Dsz);
  const v4f* ctx4 = (const v4f*)ctx;
  #pragma unroll
  for (int k = 0; k < 4; ++k) {
    const int idx = tid + k * 256;                 // row = idx>>6, col4 = idx&63
    __builtin_nontemporal_store(ctx4[idx & 63], &out4[idx]);
  }

  // attn rows: 16 x 2048 floats = 8192 v4f
  v4f* attn4 = (v4f*)(attn + ((size_t)b * Lsz + (size_t)ic * 16) * Lsz);
  #pragma unroll 8
  for (int k = 0; k < 32; ++k) {
    const int idx = tid + k * 256;                 // row = idx>>9, col4 = idx&511
    __builtin_nontemporal_store(pr4[idx & 511], &attn4[idx]);
  }
}

// ---------------------------------------------------------------------------
extern "C" void kernel_launch(void* const* d_in, const int* in_sizes, int n_in,
                              void* d_out, int out_size, void* d_ws, size_t ws_size,
                              hipStream_t stream) {
  const float* inp = (const float*)d_in[0];        // [B, L, D]
  const float* vw  = (const float*)d_in[1];        // [1, 2D]; wk = vw + D
  const float* wk  = vw + Dsz;

  float* out  = (float*)d_out;                     // [B, L, D]
  float* attn = out + (size_t)Bsz * Lsz * Dsz;     // [B, L, L]

  float* ws      = (float*)d_ws;
  float* sk      = ws;                  // 32768 floats
  float* p       = ws + 32768;          // 32768 floats
  float* partial = ws + 65536;          // 16*8*256 = 32768 floats

  sk_wmma_kernel    <<<256,          256, 0, stream>>>(inp, wk, sk);
  softmax_kernel    <<<Bsz,          256, 0, stream>>>(sk, p);
  ctx_partial_kernel<<<dim3(Bsz, 8), 256, 0, stream>>>(inp, p, partial);
  write_out_kernel  <<<Bsz * 128,    256, 0, stream>>>(p, partial, out, attn);
}